// NHNN_37795712205247
// MI455X (gfx1250) — compile-verified
//
#include <hip/hip_runtime.h>
#include <cstddef>

typedef __attribute__((ext_vector_type(2))) float v2f;
typedef __attribute__((ext_vector_type(8))) float v8f;

#define DN0 4096
#define DN1 8192
#define DN2 4096

constexpr int KSLICE = 512;  // K elements per block (grid.y slice)
constexpr int TKT    = 64;   // K per LDS tile
constexpr int TM     = 128;  // rows per block (8 waves x 16 rows)
constexpr int WSTR   = 68;   // padded LDS row stride (68 mod 64 = 4 -> conflict-free)

// ---------- x = tanh(in) (optional) + 8 Hebbian pre-sums (block partials) ----------
__global__ void pre_sums_kernel(const float* __restrict__ xin, float* __restrict__ xout,
                                const float* __restrict__ act,
                                const float* __restrict__ a, const float* __restrict__ c,
                                const float* __restrict__ d, const float* __restrict__ e,
                                float* __restrict__ partials, int n, int do_tanh)
{
    float s0 = 0.f, s1 = 0.f, s2 = 0.f, s3 = 0.f, s4 = 0.f, s5 = 0.f, s6 = 0.f, s7 = 0.f;
    for (int i = blockIdx.x * blockDim.x + threadIdx.x; i < n; i += gridDim.x * blockDim.x) {
        float x = xin[i];
        if (do_tanh) { x = tanhf(x); xout[i] = x; }
        float ai = a[i] * act[i];
        float ci = c[i] * act[i];
        float di = d[i];
        float ei = e[i];
        s0 += ei * ai * x;  s1 += ai * x;
        s2 += ei * x;       s3 += x;
        s4 += ei * ci * x;  s5 += ci * x;
        s6 += ei * di * x;  s7 += di * x;
    }
    __shared__ float red[256];
    float vals[8] = {s0, s1, s2, s3, s4, s5, s6, s7};
    for (int j = 0; j < 8; ++j) {
        red[threadIdx.x] = vals[j];
        __syncthreads();
        for (int st = 128; st > 0; st >>= 1) {
            if ((int)threadIdx.x < st) red[threadIdx.x] += red[threadIdx.x + st];
            __syncthreads();
        }
        if (threadIdx.x == 0) partials[blockIdx.x * 8 + j] = red[0];
        __syncthreads();
    }
}

__global__ void reduce8_kernel(const float* __restrict__ partials, int nb,
                               float* __restrict__ sums)
{
    int j = threadIdx.x;
    if (j < 8) {
        float s = 0.f;
        for (int k = 0; k < nb; ++k) s += partials[k * 8 + j];
        sums[j] = s;
    }
}

// ---------- GEMV via V_WMMA_F32_16X16X4_F32: partial[gy*O + o] = W[o, kslice] . x[kslice] ----------
__global__ __launch_bounds__(256) void gemv_wmma_kernel(
    const float* __restrict__ W, const float* __restrict__ x,
    float* __restrict__ partial, int K, int O)
{
    __shared__ __align__(16) float xs[KSLICE];
    __shared__ __align__(16) float wt[TM * WSTR];

    const int tid  = threadIdx.x;
    const int lane = tid & 31;
    const int wv   = tid >> 5;      // wave 0..7 -> rows [wv*16, wv*16+16)
    const int m    = lane & 15;     // matrix row within wave tile / column lane
    const int h    = lane >> 4;     // K half-select (lanes 16..31 hold K+2,K+3)

    const int row_base = blockIdx.x * TM;
    const int k_begin  = blockIdx.y * KSLICE;

    // Stage x slice (512 floats) into LDS
    if (tid < KSLICE / 4) {
        *(float4*)&xs[tid * 4] = *(const float4*)&x[k_begin + tid * 4];
    }

    v8f acc = {0.f, 0.f, 0.f, 0.f, 0.f, 0.f, 0.f, 0.f};

    for (int t = 0; t < KSLICE / TKT; ++t) {
        const int k0 = k_begin + t * TKT;
        // Cooperative coalesced load of W tile [TM x TKT] into padded LDS.
        // Per b128 instruction, 16 lanes cover one contiguous 256B row segment.
        {
            const int col = m << 2;
            const int rr  = (wv << 1) + h;
            #pragma unroll
            for (int r = 0; r < 8; ++r) {
                const int row = r * 16 + rr;
                *(float4*)&wt[row * WSTR + col] =
                    *(const float4*)&W[(size_t)(row_base + row) * K + k0 + col];
            }
        }
        __syncthreads();
        // A: 16x4 f32 tile of W rows (lane m = row, K in-lane, lanes 16+ hold K+2)
        // B: 4x16 broadcast of x chunk (all 16 columns identical)
        const float* arow = &wt[(wv * 16 + m) * WSTR + 2 * h];
        const float* brow = &xs[t * TKT + 2 * h];
        #pragma unroll
        for (int kk = 0; kk < TKT; kk += 4) {
            v2f a = *(const v2f*)(arow + kk);
            v2f b = *(const v2f*)(brow + kk);
            acc = __builtin_amdgcn_wmma_f32_16x16x4_f32(
                /*neg_a=*/false, a, /*neg_b=*/false, b,
                /*c_mod=*/(short)0, acc, /*reuse_a=*/false, /*reuse_b=*/false);
        }
        __syncthreads();
    }

    // D layout: VGPR r = rows {r, r+8}, all columns equal (B was broadcast).
    // Lane 0 holds rows 0..7, lane 16 holds rows 8..15 of this wave's tile.
    if (m == 0) {
        size_t base = (size_t)blockIdx.y * O + row_base + wv * 16 + h * 8;
        #pragma unroll
        for (int r = 0; r < 8; ++r) partial[base + r] = acc[r];
    }
}

// ---------- sum K-split partials + closed-form Hebbian term + optional tanh ----------
__global__ void heb_epilog_kernel(const float* __restrict__ partial, int G, int O,
                                  const float* __restrict__ e_post, const float* __restrict__ b_post,
                                  const float* __restrict__ act_post, const float* __restrict__ c_post,
                                  const float* __restrict__ d_post, const float* __restrict__ sums,
                                  float* __restrict__ out, int do_tanh)
{
    int o = blockIdx.x * blockDim.x + threadIdx.x;
    if (o >= O) return;
    float dot = 0.f;
    for (int g = 0; g < G; ++g) dot += partial[(size_t)g * O + o];
    float S_ae = sums[0], S_a = sums[1], S_e = sums[2], S_1 = sums[3];
    float S_ce = sums[4], S_c = sums[5], S_de = sums[6], S_d = sums[7];
    float e     = e_post[o];
    float ap    = act_post[o];
    float post  = b_post[o] * ap;
    float cpost = c_post[o] * ap;
    float heb = 0.5f * ((S_ae + e * S_a) + post * (S_e + e * S_1)
                        + cpost * (S_ce + e * S_c) + d_post[o] * (S_de + e * S_d));
    float y = dot + heb;
    out[o] = do_tanh ? tanhf(y) : y;
}

extern "C" void kernel_launch(void* const* d_in, const int* in_sizes, int n_in,
                              void* d_out, int out_size, void* d_ws, size_t ws_size,
                              hipStream_t stream)
{
    const float* inputs = (const float*)d_in[0];
    const float* act0   = (const float*)d_in[1];
    const float* act1   = (const float*)d_in[2];
    const float* act2   = (const float*)d_in[3];
    const float* W0     = (const float*)d_in[4];
    const float* W1     = (const float*)d_in[5];
    const float* a0 = (const float*)d_in[6];
    const float* c0 = (const float*)d_in[7];
    const float* d0 = (const float*)d_in[8];
    const float* e0 = (const float*)d_in[9];
    const float* a1 = (const float*)d_in[10];
    const float* b1 = (const float*)d_in[11];
    const float* c1 = (const float*)d_in[12];
    const float* d1 = (const float*)d_in[13];
    const float* e1 = (const float*)d_in[14];
    const float* b2 = (const float*)d_in[15];
    const float* c2 = (const float*)d_in[16];
    const float* d2 = (const float*)d_in[17];
    const float* e2 = (const float*)d_in[18];
    float* out = (float*)d_out;

    float* ws    = (float*)d_ws;
    float* x0    = ws;                        // 4096
    float* x1    = ws + 4096;                 // 8192
    float* py0   = ws + 12288;                // G0*N1 = 8*8192  = 65536
    float* py1   = ws + 12288 + 65536;        // G1*N2 = 16*4096 = 65536
    float* part0 = ws + 143360;               // 16*8
    float* part1 = ws + 143488;               // 32*8
    float* sums0 = ws + 143744;               // 8
    float* sums1 = ws + 143752;               // 8

    const int G0 = DN0 / KSLICE;  // 8  K-split groups for layer 0
    const int G1 = DN1 / KSLICE;  // 16 K-split groups for layer 1

    // Layer 0: x0 = tanh(inputs), 8 Hebbian pre-sums over layer-0 neurons
    pre_sums_kernel<<<16, 256, 0, stream>>>(inputs, x0, act0, a0, c0, d0, e0, part0, DN0, 1);
    reduce8_kernel<<<1, 32, 0, stream>>>(part0, 16, sums0);
    // y1 partials = W0 . x0 (K-split over grid.y)
    gemv_wmma_kernel<<<dim3(DN1 / TM, G0), 256, 0, stream>>>(W0, x0, py0, DN0, DN1);
    // x1 = tanh(sum partials + hebbian(layer0->1))
    heb_epilog_kernel<<<(DN1 + 255) / 256, 256, 0, stream>>>(py0, G0, DN1,
        e1, b1, act1, c1, d1, sums0, x1, 1);

    // Layer 1: pre-sums over layer-1 neurons (x1 already tanh'ed)
    pre_sums_kernel<<<32, 256, 0, stream>>>(x1, nullptr, act1, a1, c1, d1, e1, part1, DN1, 0);
    reduce8_kernel<<<1, 32, 0, stream>>>(part1, 32, sums1);
    // x2 partials = W1 . x1
    gemv_wmma_kernel<<<dim3(DN2 / TM, G1), 256, 0, stream>>>(W1, x1, py1, DN1, DN2);
    // out = sum partials + hebbian(layer1->2), no tanh
    heb_epilog_kernel<<<(DN2 + 255) / 256, 256, 0, stream>>>(py1, G1, DN2,
        e2, b2, act2, c2, d2, sums1, out, 0);
}